// CFM_80272938762374
// MI455X (gfx1250) — compile-verified
//
#include <hip/hip_runtime.h>

typedef float v2f __attribute__((ext_vector_type(2)));
typedef float v8f __attribute__((ext_vector_type(8)));

#define T_DIM 1024
#define O_DIM 512
#define D_DIM 256
#define H_DIM 512
#define PADH  (H_DIM + 4)   // 516 floats: rows step 4 banks -> 16 rows cover 64 banks

// ---------------------------------------------------------------------------
// Phase 1: front GEMMs with V_WMMA_F32_16X16X4_F32 (exact fp32 matrix math)
//   HT [T,H]  = z_t @ W1[0:D, :]
//   HOB[O,H]  = z_o @ W1[D:2D, :] + b1
// One wave per 16x16 output tile; K=256 in 64 steps of 4.
// ---------------------------------------------------------------------------
__global__ __launch_bounds__(128) void CFM_front_gemm(
    const float* __restrict__ z_t, const float* __restrict__ z_o,
    const float* __restrict__ W1,  const float* __restrict__ b1,
    float* __restrict__ HT, float* __restrict__ HOB)
{
  const int lane = threadIdx.x & 31;
  const int wave = (blockIdx.x << 2) | (threadIdx.x >> 5);

  const int HT_TILES = (T_DIM / 16) * (H_DIM / 16);   // 64*32 = 2048

  const float* src; const float* wbase; const float* dst_bias;
  float* dst; int rows_tile, h_tile;
  if (wave < HT_TILES) {
    rows_tile = wave >> 5;          // H/16 = 32 h-tiles per row-tile
    h_tile    = wave & 31;
    src = z_t; wbase = W1;                 dst = HT;  dst_bias = nullptr;
  } else {
    const int w2 = wave - HT_TILES;
    rows_tile = w2 >> 5;
    h_tile    = w2 & 31;
    src = z_o; wbase = W1 + D_DIM * H_DIM; dst = HOB; dst_bias = b1;
  }

  const int mrow = (rows_tile << 4) + (lane & 15);   // A: M = lane&15
  const int ncol = (h_tile    << 4) + (lane & 15);   // B/D: N = lane&15
  const int kk   = (lane >> 4) << 1;                 // lanes 16-31 hold K+2,K+3

  const float* arow = src   + mrow * D_DIM + kk;     // contiguous K pair
  const float* bcol = wbase + kk * H_DIM + ncol;     // stride-H K pair

  v8f acc = {};
  #pragma unroll 4
  for (int k0 = 0; k0 < D_DIM; k0 += 4) {
    v2f a = *(const v2f*)(arow + k0);                // K = k0+kk, k0+kk+1
    v2f b;
    b.x = bcol[k0 * H_DIM];                          // K = k0+kk
    b.y = bcol[k0 * H_DIM + H_DIM];                  // K = k0+kk+1
    acc = __builtin_amdgcn_wmma_f32_16x16x4_f32(
        /*neg_a=*/false, a, /*neg_b=*/false, b,
        /*c_mod=*/(short)0, acc, /*reuse_a=*/false, /*reuse_b=*/false);
  }

  const float bias = dst_bias ? dst_bias[ncol] : 0.0f;
  const int row0 = (rows_tile << 4) + ((lane >> 4) << 3);  // C/D: M = r + (l>>4)*8
  #pragma unroll
  for (int r = 0; r < 8; ++r) {
    dst[(row0 + r) * H_DIM + ncol] = acc[r] + bias;
  }
}

// ---------------------------------------------------------------------------
// Async global->LDS copy of one 16B chunk per lane (ASYNCcnt-tracked DMA).
// VDST operand carries the wave-relative LDS byte address; flat LDS pointers
// hold the LDS offset in their low 32 bits (aperture layout).
// ---------------------------------------------------------------------------
__device__ __forceinline__ void async_copy_b128(const float* gsrc, float* lds_dst) {
  unsigned lds_off = (unsigned)(unsigned long long)lds_dst;
  asm volatile("global_load_async_to_lds_b128 %0, %1, off"
               :
               : "v"(lds_off), "v"(gsrc)
               : "memory");
}

__device__ __forceinline__ void wait_async_all() {
  asm volatile("s_wait_asynccnt 0x0" ::: "memory");
}

// ---------------------------------------------------------------------------
// Phase 2: out[t,o] = b2 + sum_h lrelu(HT[t,h] + HOB[o,h]) * W2[h]
// 32t x 32o tile per 256-thread block, 2x2 register micro-tile per thread.
// HT/HOB tiles + W2 staged in LDS (~134KB dynamic) via async-to-LDS DMA.
// Per 4-h step: 5 ds_load_b128 feed 16 fma chains -> VALU-bound, not DS-bound.
// ---------------------------------------------------------------------------
__global__ __launch_bounds__(256) void CFM_broadcast_mlp(
    const float* __restrict__ HT, const float* __restrict__ HOB,
    const float* __restrict__ W2, const float* __restrict__ b2,
    float* __restrict__ out)
{
  extern __shared__ float smem[];
  float* s_ht = smem;                    // 32 * 516
  float* s_ho = smem + 32 * PADH;        // 32 * 516
  float* s_w2 = smem + 64 * PADH;        // 512

  const int tid = threadIdx.x;
  const int t0  = (blockIdx.x >> 4) << 5;   // O/32 = 16 o-tiles
  const int o0  = (blockIdx.x & 15) << 5;

  // Async stage-in: 32 rows x 128 float4 per tensor + 128 float4 of W2.
  for (int idx = tid; idx < 32 * (H_DIM / 4); idx += 256) {
    const int r = idx >> 7;             // / 128
    const int c = (idx & 127) << 2;
    async_copy_b128(HT  + (t0 + r) * H_DIM + c, s_ht + r * PADH + c);
    async_copy_b128(HOB + (o0 + r) * H_DIM + c, s_ho + r * PADH + c);
  }
  if (tid < H_DIM / 4) {
    async_copy_b128(W2 + (tid << 2), s_w2 + (tid << 2));
  }
  wait_async_all();
  __syncthreads();

  // 2x2 micro-tile: rows {ty, ty+16}, cols {ox, ox+16}.
  // Consecutive lanes share ty (s_ht broadcast) and have consecutive ox
  // (4-bank row stride -> conflict-free); stores coalesce over ox.
  const int ty = tid >> 4;
  const int ox = tid & 15;
  const float* htr0 = s_ht + ty * PADH;
  const float* htr1 = s_ht + (ty + 16) * PADH;
  const float* hor0 = s_ho + ox * PADH;
  const float* hor1 = s_ho + (ox + 16) * PADH;

  float acc00 = 0.0f, acc01 = 0.0f, acc10 = 0.0f, acc11 = 0.0f;
  #pragma unroll 2
  for (int h = 0; h < H_DIM; h += 4) {
    float4 a0 = *(const float4*)(htr0 + h);
    float4 a1 = *(const float4*)(htr1 + h);
    float4 b0 = *(const float4*)(hor0 + h);
    float4 b1 = *(const float4*)(hor1 + h);
    float4 w  = *(const float4*)(s_w2 + h);

    float x;
    x = a0.x + b0.x; x = fmaxf(x, 0.01f * x); acc00 = fmaf(x, w.x, acc00);
    x = a0.y + b0.y; x = fmaxf(x, 0.01f * x); acc00 = fmaf(x, w.y, acc00);
    x = a0.z + b0.z; x = fmaxf(x, 0.01f * x); acc00 = fmaf(x, w.z, acc00);
    x = a0.w + b0.w; x = fmaxf(x, 0.01f * x); acc00 = fmaf(x, w.w, acc00);

    x = a0.x + b1.x; x = fmaxf(x, 0.01f * x); acc01 = fmaf(x, w.x, acc01);
    x = a0.y + b1.y; x = fmaxf(x, 0.01f * x); acc01 = fmaf(x, w.y, acc01);
    x = a0.z + b1.z; x = fmaxf(x, 0.01f * x); acc01 = fmaf(x, w.z, acc01);
    x = a0.w + b1.w; x = fmaxf(x, 0.01f * x); acc01 = fmaf(x, w.w, acc01);

    x = a1.x + b0.x; x = fmaxf(x, 0.01f * x); acc10 = fmaf(x, w.x, acc10);
    x = a1.y + b0.y; x = fmaxf(x, 0.01f * x); acc10 = fmaf(x, w.y, acc10);
    x = a1.z + b0.z; x = fmaxf(x, 0.01f * x); acc10 = fmaf(x, w.z, acc10);
    x = a1.w + b0.w; x = fmaxf(x, 0.01f * x); acc10 = fmaf(x, w.w, acc10);

    x = a1.x + b1.x; x = fmaxf(x, 0.01f * x); acc11 = fmaf(x, w.x, acc11);
    x = a1.y + b1.y; x = fmaxf(x, 0.01f * x); acc11 = fmaf(x, w.y, acc11);
    x = a1.z + b1.z; x = fmaxf(x, 0.01f * x); acc11 = fmaf(x, w.z, acc11);
    x = a1.w + b1.w; x = fmaxf(x, 0.01f * x); acc11 = fmaf(x, w.w, acc11);
  }

  const float bb = b2[0];
  out[(t0 + ty)      * O_DIM + (o0 + ox)]      = acc00 + bb;
  out[(t0 + ty)      * O_DIM + (o0 + ox + 16)] = acc01 + bb;
  out[(t0 + ty + 16) * O_DIM + (o0 + ox)]      = acc10 + bb;
  out[(t0 + ty + 16) * O_DIM + (o0 + ox + 16)] = acc11 + bb;
}

extern "C" void kernel_launch(void* const* d_in, const int* in_sizes, int n_in,
                              void* d_out, int out_size, void* d_ws, size_t ws_size,
                              hipStream_t stream) {
  (void)in_sizes; (void)n_in; (void)out_size; (void)ws_size;
  const float* z_t = (const float*)d_in[0];
  const float* z_o = (const float*)d_in[1];
  const float* W1  = (const float*)d_in[2];
  const float* b1  = (const float*)d_in[3];
  const float* W2  = (const float*)d_in[4];
  const float* b2  = (const float*)d_in[5];

  float* HT  = (float*)d_ws;                 // T*H floats (2MB)
  float* HOB = HT + (size_t)T_DIM * H_DIM;   // O*H floats (1MB)

  // Phase 1: 3072 wave-tiles, 4 waves (128 threads) per block -> 768 blocks.
  CFM_front_gemm<<<768, 128, 0, stream>>>(z_t, z_o, W1, b1, HT, HOB);

  // Phase 2: (T/32)*(O/32) = 512 blocks, 256 threads, ~134KB dynamic LDS.
  const size_t lds_bytes = (size_t)(64 * PADH + H_DIM) * sizeof(float);
  CFM_broadcast_mlp<<<512, 256, lds_bytes, stream>>>(HT, HOB, W2, b2, (float*)d_out);
}